// RadarViT_74002286510311
// MI455X (gfx1250) — compile-verified
//
#include <hip/hip_runtime.h>
#include <math.h>

typedef __attribute__((ext_vector_type(2))) float v2f;
typedef __attribute__((ext_vector_type(8))) float v8f;

namespace {
constexpr int Bc   = 16;
constexpr int Nc   = 2048;
constexpr int Ec   = 16;
constexpr int Hc   = 4;
constexpr int DHc  = 4;
constexpr int HIDc = 64;
constexpr int FINc = 5;
constexpr int QT   = Nc / 16;     // 128 k/q tiles per (batch, head)
constexpr int TOK  = Bc * Nc;     // 32768 tokens
}

// Exchange a value with the partner lane in the other 16-lane half (lane ^ 16).
// v_permlanex16_b32 is a VALU cross-row permute: far cheaper than ds_bpermute.
__device__ __forceinline__ float swap16(float x) {
#if __has_builtin(__builtin_amdgcn_permlanex16)
  unsigned u = __float_as_uint(x);
  unsigned r = __builtin_amdgcn_permlanex16(u, u, 0x76543210u, 0xfedcba98u, false, false);
  return __uint_as_float(r);
#else
  return __shfl_xor(x, 16, 32);
#endif
}

// ---------------- Kernel 1: x @ Wp^T + b -> h ; LN1 ; qkv projection ----------------
__global__ void rvit_embed_qkv(const float* __restrict__ x,
                               const float* __restrict__ Wp,  const float* __restrict__ bp,
                               const float* __restrict__ g1,  const float* __restrict__ be1,
                               const float* __restrict__ Win, const float* __restrict__ bin,
                               float* __restrict__ h_ws,
                               float* __restrict__ q_ws, float* __restrict__ k_ws,
                               float* __restrict__ v_ws) {
  int t = blockIdx.x * blockDim.x + threadIdx.x;
  if (t >= TOK) return;
  int b = t / Nc, n = t % Nc;

  float xin[FINc];
#pragma unroll
  for (int f = 0; f < FINc; ++f) xin[f] = x[t * FINc + f];

  float h[Ec];
#pragma unroll
  for (int j = 0; j < Ec; ++j) {
    float a = bp[j];
#pragma unroll
    for (int f = 0; f < FINc; ++f) a = fmaf(xin[f], Wp[j * FINc + f], a);
    h[j] = a;
    h_ws[t * Ec + j] = a;
  }

  float mu = 0.f;
#pragma unroll
  for (int j = 0; j < Ec; ++j) mu += h[j];
  mu *= (1.f / Ec);
  float var = 0.f;
#pragma unroll
  for (int j = 0; j < Ec; ++j) { float d = h[j] - mu; var = fmaf(d, d, var); }
  var *= (1.f / Ec);
  float rs = rsqrtf(var + 1e-5f);
  float hn[Ec];
#pragma unroll
  for (int j = 0; j < Ec; ++j) hn[j] = (h[j] - mu) * rs * g1[j] + be1[j];

  // qkv = hn @ Win^T + bin; q pre-scaled by 1/sqrt(DH)=0.5; layout [B][H][N][DH]
#pragma unroll
  for (int j = 0; j < 3 * Ec; ++j) {
    float a = bin[j];
#pragma unroll
    for (int i = 0; i < Ec; ++i) a = fmaf(hn[i], Win[j * Ec + i], a);
    int jj = j & 15;
    int hd = jj >> 2, d = jj & 3;
    long idx = (((long)b * Hc + hd) * Nc + n) * DHc + d;
    if (j < Ec)            q_ws[idx] = a * 0.5f;
    else if (j < 2 * Ec)   k_ws[idx] = a;
    else                   v_ws[idx] = a;
  }
}

// ---------------- Kernel 2: WMMA flash attention -------------------------------------
// One wave per (batch, head, 16-row q tile). Scores computed TRANSPOSED:
//   S^T[key][q] = K_tile[16,4] x Q^T[4,16]  (one V_WMMA_F32_16X16X4_F32)
// so each lane holds 8 keys of one q column -> softmax reductions are in-lane
// trees + a single permlanex16 cross-half combine. P.V done as
//   O^T[dh,q] = V^T x P^T  (4 chained WMMAs), with P^T operands built from
// registers via permlanex16 exchange — no LDS anywhere.
__global__ __launch_bounds__(256) void rvit_attn(const float* __restrict__ q_ws,
                                                 const float* __restrict__ k_ws,
                                                 const float* __restrict__ v_ws,
                                                 float* __restrict__ o_ws) {
  const int lane = threadIdx.x & 31;
  const int col  = lane & 15;            // q column (and A-row / B-col index)
  const int half = lane >> 4;

  const int unit = blockIdx.x * 8 + (threadIdx.x >> 5);  // 8192 units
  const int b    = unit / (Hc * QT);
  const int rem  = unit % (Hc * QT);
  const int hd   = rem / QT;
  const int qt   = rem % QT;

  const float* qh = q_ws + (((long)b * Hc + hd) * Nc + qt * 16) * DHc;
  const float* kb = k_ws + (((long)b * Hc + hd) * Nc) * DHc;
  const float* vb = v_ws + (((long)b * Hc + hd) * Nc) * DHc;

  // B operand = Q^T [4 x 16] (loop invariant): lane holds col q=col, K-rows {0,1}+2*half
  v2f Bq;
  Bq.x = qh[col * DHc + 2 * half + 0];
  Bq.y = qh[col * DHc + 2 * half + 1];

  v8f acc = {0.f, 0.f, 0.f, 0.f, 0.f, 0.f, 0.f, 0.f};  // O^T: rows dh (0..3 valid), cols q
  float m = -3.0e38f, l = 0.f;                         // flash state for q=col

  for (int kt = 0; kt < QT; ++kt) {
    const float* kh = kb + kt * 16 * DHc;
    const float* vh = vb + kt * 16 * DHc;

    // A operand = K tile [16 keys x 4]: lane holds row M=col(key), K = 2*half+{0,1}
    v2f Ak;
    Ak.x = kh[col * DHc + 2 * half + 0];
    Ak.y = kh[col * DHc + 2 * half + 1];

    v8f z = {0.f, 0.f, 0.f, 0.f, 0.f, 0.f, 0.f, 0.f};
    v8f S = __builtin_amdgcn_wmma_f32_16x16x4_f32(false, Ak, false, Bq,
                                                  (short)0, z, false, false);
    // lane elem e = S^T[key = e + 8*half][q = col]

    // column (q) max: in-lane tree over 8 keys + one cross-half combine
    float a0 = fmaxf(S[0], S[1]), a1 = fmaxf(S[2], S[3]);
    float a2 = fmaxf(S[4], S[5]), a3 = fmaxf(S[6], S[7]);
    float pm = fmaxf(fmaxf(a0, a1), fmaxf(a2, a3));
    pm = fmaxf(pm, swap16(pm));
    float mnew = fmaxf(m, pm);
    float corr = __expf(m - mnew);
    m = mnew;

    float p[8];
#pragma unroll
    for (int e = 0; e < 8; ++e) p[e] = __expf(S[e] - mnew);

    float sum = ((p[0] + p[1]) + (p[2] + p[3])) + ((p[4] + p[5]) + (p[6] + p[7]));
    sum += swap16(sum);
    l = l * corr + sum;

    // rescale the live rows of O^T (rows 0..3 live in half-0 elems 0..3)
    acc[0] *= corr; acc[1] *= corr; acc[2] *= corr; acc[3] *= corr;

    // partner-half P^T values for B-operand assembly
    float o0 = swap16(p[0]), o1 = swap16(p[1]), o2 = swap16(p[2]), o3 = swap16(p[3]);
    float o4 = swap16(p[4]), o5 = swap16(p[5]), o6 = swap16(p[6]), o7 = swap16(p[7]);

    // O^T += V^T[:,4j..4j+3] x P^T[4j..4j+3,:]  for j = 0..3
    // A = V^T chunk: lane row m=col(dh<4), k=2*half+{0,1} -> V[4j+2*half+{0,1}][col]
    // B = P^T chunk: lane col q=col, k-row = {0,1}+2*half -> key 4j+2*half+{0,1}
    {
      v2f Av, Bp;
      Av.x = (col < DHc) ? vh[(0 + 2 * half) * DHc + col] : 0.f;
      Av.y = (col < DHc) ? vh[(1 + 2 * half) * DHc + col] : 0.f;
      Bp.x = half ? o2 : p[0];  Bp.y = half ? o3 : p[1];
      acc = __builtin_amdgcn_wmma_f32_16x16x4_f32(false, Av, false, Bp,
                                                  (short)0, acc, false, false);
    }
    {
      v2f Av, Bp;
      Av.x = (col < DHc) ? vh[(4 + 2 * half) * DHc + col] : 0.f;
      Av.y = (col < DHc) ? vh[(5 + 2 * half) * DHc + col] : 0.f;
      Bp.x = half ? o6 : p[4];  Bp.y = half ? o7 : p[5];
      acc = __builtin_amdgcn_wmma_f32_16x16x4_f32(false, Av, false, Bp,
                                                  (short)0, acc, false, false);
    }
    {
      v2f Av, Bp;
      Av.x = (col < DHc) ? vh[(8 + 2 * half) * DHc + col] : 0.f;
      Av.y = (col < DHc) ? vh[(9 + 2 * half) * DHc + col] : 0.f;
      Bp.x = half ? p[2] : o0;  Bp.y = half ? p[3] : o1;
      acc = __builtin_amdgcn_wmma_f32_16x16x4_f32(false, Av, false, Bp,
                                                  (short)0, acc, false, false);
    }
    {
      v2f Av, Bp;
      Av.x = (col < DHc) ? vh[(12 + 2 * half) * DHc + col] : 0.f;
      Av.y = (col < DHc) ? vh[(13 + 2 * half) * DHc + col] : 0.f;
      Bp.x = half ? p[6] : o4;  Bp.y = half ? p[7] : o5;
      acc = __builtin_amdgcn_wmma_f32_16x16x4_f32(false, Av, false, Bp,
                                                  (short)0, acc, false, false);
    }
  }

  // O^T rows 0..3 (dh) live entirely in half-0 lanes, cols = q = col.
  if (half == 0) {
    float inv = 1.f / l;
    float* orow = o_ws + ((long)b * Nc + qt * 16 + col) * Ec + hd * DHc;
    orow[0] = acc[0] * inv;
    orow[1] = acc[1] * inv;
    orow[2] = acc[2] * inv;
    orow[3] = acc[3] * inv;
  }
}

// ---------------- Kernel 3: out_proj + residual, LN2, exact-GELU MLP + residual -----
__global__ void rvit_post(const float* __restrict__ h_ws, const float* __restrict__ o_ws,
                          const float* __restrict__ Wout, const float* __restrict__ bout,
                          const float* __restrict__ g2,   const float* __restrict__ be2,
                          const float* __restrict__ W1,   const float* __restrict__ bb1,
                          const float* __restrict__ W2,   const float* __restrict__ bb2,
                          float* __restrict__ out) {
  int t = blockIdx.x * blockDim.x + threadIdx.x;
  if (t >= TOK) return;

  float o[Ec], h[Ec];
#pragma unroll
  for (int i = 0; i < Ec; ++i) { o[i] = o_ws[t * Ec + i]; h[i] = h_ws[t * Ec + i]; }

#pragma unroll
  for (int j = 0; j < Ec; ++j) {
    float a = bout[j];
#pragma unroll
    for (int i = 0; i < Ec; ++i) a = fmaf(o[i], Wout[j * Ec + i], a);
    h[j] += a;
  }

  float mu = 0.f;
#pragma unroll
  for (int j = 0; j < Ec; ++j) mu += h[j];
  mu *= (1.f / Ec);
  float var = 0.f;
#pragma unroll
  for (int j = 0; j < Ec; ++j) { float d = h[j] - mu; var = fmaf(d, d, var); }
  var *= (1.f / Ec);
  float rs = rsqrtf(var + 1e-5f);
  float hn[Ec];
#pragma unroll
  for (int j = 0; j < Ec; ++j) hn[j] = (h[j] - mu) * rs * g2[j] + be2[j];

  float accj[Ec];
#pragma unroll
  for (int j = 0; j < Ec; ++j) accj[j] = bb2[j];
  for (int kk = 0; kk < HIDc; ++kk) {
    float u = bb1[kk];
#pragma unroll
    for (int i = 0; i < Ec; ++i) u = fmaf(hn[i], W1[kk * Ec + i], u);
    float g = 0.5f * u * (1.f + erff(u * 0.70710678118654752f));  // exact GELU
#pragma unroll
    for (int j = 0; j < Ec; ++j) accj[j] = fmaf(g, W2[j * HIDc + kk], accj[j]);
  }

#pragma unroll
  for (int j = 0; j < Ec; ++j) out[t * Ec + j] = h[j] + accj[j];
}

extern "C" void kernel_launch(void* const* d_in, const int* in_sizes, int n_in,
                              void* d_out, int out_size, void* d_ws, size_t ws_size,
                              hipStream_t stream) {
  const float* x    = (const float*)d_in[0];
  const float* Wp   = (const float*)d_in[1];
  const float* bp   = (const float*)d_in[2];
  const float* g1   = (const float*)d_in[3];
  const float* be1  = (const float*)d_in[4];
  const float* Win  = (const float*)d_in[5];
  const float* bin  = (const float*)d_in[6];
  const float* Wout = (const float*)d_in[7];
  const float* bout = (const float*)d_in[8];
  const float* g2   = (const float*)d_in[9];
  const float* be2  = (const float*)d_in[10];
  const float* W1   = (const float*)d_in[11];
  const float* bb1  = (const float*)d_in[12];
  const float* W2   = (const float*)d_in[13];
  const float* bb2  = (const float*)d_in[14];
  float* out = (float*)d_out;

  float* ws   = (float*)d_ws;
  float* h_ws = ws;                       // TOK*16
  float* q_ws = h_ws + (size_t)TOK * Ec;  // [B][H][N][DH]
  float* k_ws = q_ws + (size_t)TOK * Ec;
  float* v_ws = k_ws + (size_t)TOK * Ec;
  float* o_ws = v_ws + (size_t)TOK * Ec;  // total 10 MB of scratch

  rvit_embed_qkv<<<(TOK + 255) / 256, 256, 0, stream>>>(
      x, Wp, bp, g1, be1, Win, bin, h_ws, q_ws, k_ws, v_ws);

  rvit_attn<<<(Bc * Hc * QT) / 8, 256, 0, stream>>>(q_ws, k_ws, v_ws, o_ws);

  rvit_post<<<(TOK + 255) / 256, 256, 0, stream>>>(
      h_ws, o_ws, Wout, bout, g2, be2, W1, bb1, W2, bb2, out);
}